// LSTM_Layer_27788438405737
// MI455X (gfx1250) — compile-verified
//
#include <hip/hip_runtime.h>
#include <hip/hip_bf16.h>
#include <math.h>

// Problem constants (from reference setup_inputs)
constexpr int H = 8;     // LSTM units
constexpr int D = 32;    // input feature dim
constexpr int B = 256;
constexpr int T = 2048;
constexpr int G4 = 4 * H;  // 32 gate columns

typedef float v2f __attribute__((ext_vector_type(2)));
typedef float v8f __attribute__((ext_vector_type(8)));

// ---------------------------------------------------------------------------
// Kernel 1: Zx = x @ Wx + b, using V_WMMA_F32_16X16X4_F32.
//   x  : [B*T, 32] row-major (row = b*T + t)
//   Wx : [32, 32]  row-major (col = gate*8 + j)
//   zx : [B*T, 32] stored remapped as [row][j*4 + gate] so the recurrence
//        lane (j) reads its 4 gate pre-activations as one contiguous float4.
//
// Per wave: 16 rows (M=16), K=32 done as 8 chunks of K=4, N=32 as two
// 16-wide tiles -> 16 v_wmma_f32_16x16x4_f32 per wave.
//
// f32 WMMA VGPR layouts (ISA 7.12.2):
//   A 16x4:  VGPR r, lanes 0-15: M=lane, K=r;   lanes 16-31: M=lane-16, K=r+2
//   B 4x16:  VGPR r, lanes 0-15: K=r, N=lane;   lanes 16-31: K=r+2, N=lane-16
//   C 16x16: VGPR r, lanes 0-15: M=r, N=lane;   lanes 16-31: M=r+8, N=lane-16
// ---------------------------------------------------------------------------
__global__ void lstm_zx_wmma(const float* __restrict__ x,
                             const float* __restrict__ Wx,
                             const float* __restrict__ bias,
                             float* __restrict__ zx) {
  const int lane = threadIdx.x & 31;
  const int half = lane >> 4;    // 0: lanes 0-15, 1: lanes 16-31
  const int l16  = lane & 15;
  const long waveId  = (long)blockIdx.x * (blockDim.x >> 5) + (threadIdx.x >> 5);
  const long rowBase = waveId * 16;            // 16 rows of (b,t) per wave
  if (rowBase >= (long)B * T) return;

  // Preload B-matrix fragments (Wx is tiny: 4 KB, L2/WGP$ resident).
  // bfrag[nt][kk]: N-tile nt, K-chunk kk; .x = row kbase+2*half, .y = +1
  v2f bfrag[2][8];
#pragma unroll
  for (int nt = 0; nt < 2; ++nt) {
#pragma unroll
    for (int kk = 0; kk < 8; ++kk) {
      const int kb  = kk * 4 + 2 * half;
      const int col = nt * 16 + l16;
      v2f bf;
      bf.x = Wx[(kb + 0) * G4 + col];
      bf.y = Wx[(kb + 1) * G4 + col];
      bfrag[nt][kk] = bf;
    }
  }

  v8f acc0 = {};  // N-tile 0
  v8f acc1 = {};  // N-tile 1

  const float* arow = x + (rowBase + l16) * D;
#pragma unroll
  for (int kk = 0; kk < 8; ++kk) {
    const int kb = kk * 4 + 2 * half;
    v2f a;
    a.x = arow[kb + 0];
    a.y = arow[kb + 1];
    acc0 = __builtin_amdgcn_wmma_f32_16x16x4_f32(
        false, a, false, bfrag[0][kk], (short)0, acc0, false, false);
    acc1 = __builtin_amdgcn_wmma_f32_16x16x4_f32(
        false, a, false, bfrag[1][kk], (short)0, acc1, false, false);
  }

  // Store D with gate remap + bias: n = gate*8+j  ->  offset j*4+gate
#pragma unroll
  for (int nt = 0; nt < 2; ++nt) {
    const int n    = nt * 16 + l16;
    const int gate = n >> 3;
    const int j    = n & 7;
    const float bv = bias[n];
    const v8f& acc = nt ? acc1 : acc0;
#pragma unroll
    for (int r = 0; r < 8; ++r) {
      const long row = rowBase + r + 8 * half;
      zx[row * G4 + j * 4 + gate] = acc[r] + bv;
    }
  }
}

// ---------------------------------------------------------------------------
// Kernel 2: sequential recurrence. One wave32 per block, 4 sequences/wave,
// lane = g*8 + j  (g: local batch row 0..3, j: hidden unit 0..7).
// h is exchanged within the 8-lane group via ds_bpermute (__shfl), no
// barriers needed (wave-synchronous). zx float4 per step double-buffered in
// registers, CH timesteps per chunk, so L2 latency is off the serial chain.
// ---------------------------------------------------------------------------
__device__ __forceinline__ float sig_f(float x) {
  return 1.0f / (1.0f + __expf(-x));          // x->-inf: exp->inf -> 0  (safe)
}
__device__ __forceinline__ float tanh_f(float x) {
  const float e = __expf(-2.0f * x);          // overflow-safe both directions
  return 2.0f / (1.0f + e) - 1.0f;
}

template <bool USE_WS>
__global__ void lstm_recurrence(const float* __restrict__ zx,   // USE_WS path
                                const float* __restrict__ x,    // fused path
                                const float* __restrict__ Wx,
                                const float* __restrict__ Wh,   // [8][32]
                                const float* __restrict__ bias, // [32]
                                const float* __restrict__ h0,
                                const float* __restrict__ c0,
                                float* __restrict__ out,        // [B,T,8]
                                float* __restrict__ hf,
                                float* __restrict__ cf) {
  const int lane = threadIdx.x & 31;
  const int j    = lane & 7;
  const int grpBase = lane & 24;               // first lane of 8-lane group
  const int b    = blockIdx.x * 4 + (lane >> 3);

  // Wh columns for this lane's hidden unit, all 4 gates.
  float wh_i[H], wh_f[H], wh_g[H], wh_o[H];
#pragma unroll
  for (int k = 0; k < H; ++k) {
    const float* wr = Wh + k * G4;
    wh_i[k] = wr[0 * H + j];
    wh_f[k] = wr[1 * H + j];
    wh_g[k] = wr[2 * H + j];
    wh_o[k] = wr[3 * H + j];
  }

  // Fused-path weights (only live when !USE_WS).
  float wx_i[D], wx_f[D], wx_g[D], wx_o[D];
  float4 bz = {0.f, 0.f, 0.f, 0.f};
  if (!USE_WS) {
#pragma unroll
    for (int d = 0; d < D; ++d) {
      const float* wr = Wx + d * G4;
      wx_i[d] = wr[0 * H + j];
      wx_f[d] = wr[1 * H + j];
      wx_g[d] = wr[2 * H + j];
      wx_o[d] = wr[3 * H + j];
    }
    bz.x = bias[0 * H + j];
    bz.y = bias[1 * H + j];
    bz.z = bias[2 * H + j];
    bz.w = bias[3 * H + j];
  }

  float h = h0[b * H + j];
  float c = c0[b * H + j];

  // zx row for this sequence, viewed as float4 per (t): offset j*4 floats.
  const float4* zp = USE_WS
      ? reinterpret_cast<const float4*>(zx + (size_t)b * T * G4 + j * 4)
      : nullptr;                                // stride 8 float4 per t
  const float* xrow = x + (size_t)b * T * D;

  constexpr int CH = 8;                         // timesteps per register chunk
  float4 buf[2][CH];

  auto load_chunk = [&](int t0, int s) {
#pragma unroll
    for (int tt = 0; tt < CH; ++tt) buf[s][tt] = zp[(size_t)(t0 + tt) * 8];
  };

  if (USE_WS) load_chunk(0, 0);

  float* orow = out + ((size_t)b * T) * H + j;

  for (int t0 = 0; t0 < T; t0 += CH) {
    const int cur = (t0 / CH) & 1;
    if (USE_WS && (t0 + CH < T)) load_chunk(t0 + CH, cur ^ 1);  // prefetch

#pragma unroll
    for (int tt = 0; tt < CH; ++tt) {
      float4 z;
      if (USE_WS) {
        z = buf[cur][tt];
      } else {
        // Fused fallback: z = bias + xt @ Wx (per-lane dot products)
        z = bz;
        const float* xt = xrow + (size_t)(t0 + tt) * D;
#pragma unroll
        for (int d = 0; d < D; ++d) {
          const float xv = xt[d];
          z.x = fmaf(xv, wx_i[d], z.x);
          z.y = fmaf(xv, wx_f[d], z.y);
          z.z = fmaf(xv, wx_g[d], z.z);
          z.w = fmaf(xv, wx_o[d], z.w);
        }
      }

      // z += h @ Wh : gather the group's 8 h values via lane permute.
#pragma unroll
      for (int k = 0; k < H; ++k) {
        const float hk = __shfl(h, grpBase + k, 32);
        z.x = fmaf(hk, wh_i[k], z.x);
        z.y = fmaf(hk, wh_f[k], z.y);
        z.z = fmaf(hk, wh_g[k], z.z);
        z.w = fmaf(hk, wh_o[k], z.w);
      }

      const float ig = sig_f(z.x);
      const float fg = sig_f(z.y);
      const float gg = tanh_f(z.z);
      const float og = sig_f(z.w);
      c = fmaf(fg, c, ig * gg);
      h = og * tanh_f(c);

      orow[(size_t)(t0 + tt) * H] = h;
    }
  }

  hf[b * H + j] = h;
  cf[b * H + j] = c;
}

// ---------------------------------------------------------------------------
extern "C" void kernel_launch(void* const* d_in, const int* in_sizes, int n_in,
                              void* d_out, int out_size, void* d_ws, size_t ws_size,
                              hipStream_t stream) {
  const float* x    = (const float*)d_in[0];  // [B,T,D]
  const float* h0   = (const float*)d_in[1];  // [B,H]
  const float* c0   = (const float*)d_in[2];  // [B,H]
  const float* Wx   = (const float*)d_in[3];  // [D,4H]
  const float* Wh   = (const float*)d_in[4];  // [H,4H]
  const float* bias = (const float*)d_in[5];  // [4H]

  float* out = (float*)d_out;                         // [B,T,H]
  float* hf  = out + (size_t)B * T * H;               // [B,H]
  float* cf  = hf + (size_t)B * H;                    // [B,H]

  const size_t zx_bytes = (size_t)B * T * G4 * sizeof(float);  // 64 MB

  if (ws_size >= zx_bytes) {
    float* zx = (float*)d_ws;
    // Phase 1: WMMA input projection. 16 rows/wave, 8 waves/block.
    const long rows = (long)B * T;                     // 524288
    const int waves = (int)(rows / 16);                // 32768
    const int blocks = waves / 8;                      // 4096
    lstm_zx_wmma<<<blocks, 256, 0, stream>>>(x, Wx, bias, zx);
    // Phase 2: latency-optimized recurrence, 1 wave / block, 4 seqs / wave.
    lstm_recurrence<true><<<B / 4, 32, 0, stream>>>(zx, x, Wx, Wh, bias,
                                                    h0, c0, out, hf, cf);
  } else {
    // Fused fallback (no workspace): recompute x@Wx inside the recurrence.
    lstm_recurrence<false><<<B / 4, 32, 0, stream>>>(nullptr, x, Wx, Wh, bias,
                                                     h0, c0, out, hf, cf);
  }
}